// DeepBSpline_9371618640348
// MI455X (gfx1250) — compile-verified
//
#include <hip/hip_runtime.h>
#include <stdint.h>

// DeepBSpline activation, MI455X (gfx1250).
// x: (16,64,256,256) f32, coefficients_vect: (64*51,) f32 -> out f32 same shape.
// Pure HBM-bound elementwise op: 8 B/element -> 537 MB/call -> ~23 us @ 23.3 TB/s.
// WMMA is inapplicable (data-dependent gather); the gfx1250-optimal shape is:
//   - block-uniform channel, coefficient table staged via async global->LDS DMA
//   - pair table in LDS: one ds_load_b64 gather per element
//   - b128 non-temporal vector loads/stores, wave32, minimal VALU per element

typedef float v4f __attribute__((ext_vector_type(4)));

#define KSIZE 51
#define KHALF 25
#define NCH   64
#define GRIDF 0.16f
#define INV_GRID (1.0f / GRIDF)     // folds to exactly 6.25f
#define HW    65536                 // 256*256 elements per (n,c) plane
#define BLOCK 256
#define CHUNK_F4 2048               // float4s per block
#define BLOCKS_PER_PLANE 8          // 16384 / 2048
#define ITERS 8                     // 2048 / 256

__global__ __launch_bounds__(BLOCK) void deep_bspline_kernel(
    const float* __restrict__ x,
    const float* __restrict__ cv,
    float* __restrict__ out)
{
    __shared__ float  raw[KSIZE];        // channel's 51 knot coefficients
    __shared__ float2 pairs[KSIZE - 1];  // (cv[k], cv[k+1]) for k in [0,49]
    __shared__ float  slopes[2];         // left / right extrapolation slopes

    const int tid  = threadIdx.x;
    const int c    = blockIdx.y & (NCH - 1);   // plane = n*64 + c
    const int base = c * KSIZE;

    // Stage the 51 coefficients into LDS with the CDNA5 async DMA path
    // (global -> LDS, no VGPR round trip, tracked by ASYNCcnt).
    if (tid < KSIZE) {
        unsigned           lds_off = (unsigned)(size_t)(&raw[tid]);  // LDS byte offset
        unsigned long long gaddr   = (unsigned long long)(const void*)(cv + base + tid);
        asm volatile("global_load_async_to_lds_b32 %0, %1, off"
                     :: "v"(lds_off), "v"(gaddr) : "memory");
    }
    asm volatile("s_wait_asynccnt 0" ::: "memory");
    __syncthreads();

    // Build the 8-byte-aligned pair table: one ds_load_b64 per element later.
    if (tid < KSIZE - 1) {
        pairs[tid] = make_float2(raw[tid], raw[tid + 1]);
    }
    if (tid == 0) {
        slopes[0] = (raw[1] - raw[0]) * INV_GRID;                   // left slope
        slopes[1] = (raw[KSIZE - 1] - raw[KSIZE - 2]) * INV_GRID;   // right slope
    }
    __syncthreads();

    const float lsl  = slopes[0];
    const float rsl  = slopes[1];
    const float xmin = -GRIDF * (float)KHALF;        // -4.0
    const float xmax =  GRIDF * (float)(KHALF - 1);  //  3.84

    const size_t planeBase = (size_t)blockIdx.y * (size_t)HW;
    const size_t chunkBase = (size_t)blockIdx.x * (size_t)(CHUNK_F4 * 4);
    const v4f* __restrict__ xin = (const v4f*)(x   + planeBase + chunkBase);
    v4f*       __restrict__ xo  = (v4f*)      (out + planeBase + chunkBase);

    #pragma unroll
    for (int it = 0; it < ITERS; ++it) {
        const int i4 = it * BLOCK + tid;
        const v4f v = __builtin_nontemporal_load(xin + i4);   // global_load_b128 (NT)
        v4f r;
        #pragma unroll
        for (int j = 0; j < 4; ++j) {
            const float xv = v[j];
            // clamp + bin (fma bias: u >= 0 so trunc == floor), k in [0,49]
            const float xc = fminf(fmaxf(xv, xmin), xmax);
            const float u  = __builtin_fmaf(xc, INV_GRID, (float)KHALF);
            const int   k  = (int)u;
            const float fr = u - (float)k;
            const float2 p = pairs[k];                        // one ds_load_b64
            // lerp between the two knots
            float o = __builtin_fmaf(fr, p.y - p.x, p.x);
            // both extrapolation tails in one fused term:
            // d = xv - xc == min(xv-xmin,0) + max(xv-xmax,0); only one is nonzero
            const float d  = xv - xc;
            const float sl = (d < 0.0f) ? lsl : rsl;
            o = __builtin_fmaf(d, sl, o);
            r[j] = o;
        }
        __builtin_nontemporal_store(r, xo + i4);              // global_store_b128 (NT)
    }
}

extern "C" void kernel_launch(void* const* d_in, const int* in_sizes, int n_in,
                              void* d_out, int out_size, void* d_ws, size_t ws_size,
                              hipStream_t stream) {
    const float* x  = (const float*)d_in[0];  // (16,64,256,256) f32
    const float* cv = (const float*)d_in[1];  // (64*51,) f32
    // d_in[2] is the python scalar `size` (== 51, baked in at compile time).
    float* out = (float*)d_out;

    const int planes = out_size / HW;         // 16*64 = 1024
    dim3 grid(BLOCKS_PER_PLANE, planes);
    dim3 block(BLOCK);
    hipLaunchKernelGGL(deep_bspline_kernel, grid, block, 0, stream, x, cv, out);
}